// GCNConv_50757923504131
// MI455X (gfx1250) — compile-verified
//
#include <hip/hip_runtime.h>

// ---------------------------------------------------------------------------
// GCNConv for MI455X (gfx1250, wave32).
//   Stage 1: XW = X @ W  via V_WMMA_F32_16X16X4_F32 (fp32 in, fp32 acc:
//            exact w.r.t. the fp32 reference; problem is HBM-bound so fp32
//            WMMA rate is irrelevant).
//   Stage 2: out[i] = sum_{j in N(i)} XW[j]  (CSR gather-sum; XW fits in the
//            192MB L2 so the 205MB of gather reads stay on-chip).
// ---------------------------------------------------------------------------

typedef __attribute__((ext_vector_type(2))) float v2f;
typedef __attribute__((ext_vector_type(8))) float v8f;

#define GCN_N_NODES 50000
#define GCN_D_IN    128
#define GCN_D_OUT   64
#define GCN_M_TILES (GCN_N_NODES / 16)   // 3125 (exact)
#define GCN_N_TILES (GCN_D_OUT / 16)     // 4
#define GCN_TILES   (GCN_M_TILES * GCN_N_TILES)  // 12500 waves

// ---------------------------------------------------------------------------
// Stage 1: dense GEMM, one wave32 per 16x16 tile of XW.
// ---------------------------------------------------------------------------
__global__ __launch_bounds__(256) void gcn_gemm_wmma(
    const float* __restrict__ X,      // [50000,128] row-major
    const float* __restrict__ W,      // [128,64]    row-major
    float* __restrict__ XW)           // [50000,64]  row-major
{
    // Stage W into LDS once per block (128*64*4 = 32 KB of the 320 KB WGP LDS).
    __shared__ float Wlds[GCN_D_IN * GCN_D_OUT];
    {
        const float4* src = (const float4*)W;
        float4*       dst = (float4*)Wlds;
        const int nvec = GCN_D_IN * GCN_D_OUT / 4;   // 2048 float4
        for (int i = threadIdx.x; i < nvec; i += blockDim.x)
            dst[i] = src[i];
    }
    __syncthreads();

    const int lane = threadIdx.x & 31;
    const int wave = blockIdx.x * 8 + (threadIdx.x >> 5);
    if (wave >= GCN_TILES) return;          // wave-uniform: EXEC stays all-1s

    const int tileN = wave & (GCN_N_TILES - 1);
    const int tileM = wave >> 2;

    // --- WMMA f32 16x16x4 operand layouts (ISA 7.12.2, wave32) ---
    // A (16x4, MxK): lanes 0-15 -> M=lane,  VGPR0/1 = K0/K1
    //               lanes 16-31 -> M=lane-16, VGPR0/1 = K2/K3
    // B (4x16, KxN): lanes 0-15 -> N=lane,  VGPR0/1 = rows K0/K1
    //               lanes 16-31 -> N=lane-16, VGPR0/1 = rows K2/K3
    // C/D (16x16):  VGPR r -> M = r (lanes 0-15) / r+8 (lanes 16-31), N = lane&15
    const int mn   = lane & 15;             // M-in-tile for A, N-in-tile for B/C/D
    const int koff = (lane >> 4) << 1;      // 0 or 2

    const float* Xrow = X + (size_t)(tileM * 16 + mn) * GCN_D_IN;
    const float* Wcol = Wlds + (tileN * 16 + mn);

    __builtin_prefetch(Xrow + 64, 0, 0);    // global_prefetch_b8 on the row tail

    v8f acc = {};
#pragma unroll
    for (int kk = 0; kk < GCN_D_IN; kk += 4) {
        v2f a, b;
        a.x = Xrow[kk + koff];
        a.y = Xrow[kk + koff + 1];
        b.x = Wcol[(kk + koff)     * GCN_D_OUT];
        b.y = Wcol[(kk + koff + 1) * GCN_D_OUT];
        // 8 args: (neg_a, A, neg_b, B, c_mod, C, reuse_a, reuse_b)
        acc = __builtin_amdgcn_wmma_f32_16x16x4_f32(
            false, a, false, b, (short)0, acc, false, false);
    }

    const int crow0 = tileM * 16 + ((lane >> 4) << 3);   // M of acc[0]
    float* dst = XW + (size_t)crow0 * GCN_D_OUT + tileN * 16 + mn;
#pragma unroll
    for (int r = 0; r < 8; ++r)
        dst[(size_t)r * GCN_D_OUT] = acc[r];
}

// ---------------------------------------------------------------------------
// Stage 2: CSR gather-sum. One wave per node; each lane owns 2 columns.
// Per neighbor: one coalesced 256B float2 load from XW (L2-resident).
// ---------------------------------------------------------------------------
__global__ __launch_bounds__(256) void gcn_spmm_agg(
    const float* __restrict__ XW,            // [50000,64]
    const int*   __restrict__ nodePtr,       // [50001]
    const int*   __restrict__ edgeList,      // [800000]
    float*       __restrict__ out)           // [50000,64]
{
    const int lane = threadIdx.x & 31;
    const int node = (blockIdx.x * blockDim.x + threadIdx.x) >> 5;
    if (node >= GCN_N_NODES) return;

    const int beg = nodePtr[node];
    const int end = nodePtr[node + 1];

    const float2* XW2 = (const float2*)XW;   // 32 float2 per row
    float2 acc = make_float2(0.f, 0.f);
    for (int e = beg; e < end; ++e) {
        const int j = edgeList[e];
        const float2 v = XW2[(size_t)j * 32 + lane];
        acc.x += v.x;
        acc.y += v.y;
    }
    ((float2*)out)[(size_t)node * 32 + lane] = acc;
}

// ---------------------------------------------------------------------------
// Launch. Inputs (setup_inputs order): 0=X, 1=weights, 2=nodePointer,
// 3=edgeList, 4..=packing metadata (unused in the math).
// d_ws holds the XW intermediate (50000*64*4 = 12.8 MB).
// ---------------------------------------------------------------------------
extern "C" void kernel_launch(void* const* d_in, const int* in_sizes, int n_in,
                              void* d_out, int out_size, void* d_ws, size_t ws_size,
                              hipStream_t stream) {
    (void)in_sizes; (void)n_in; (void)out_size; (void)ws_size;

    const float* X        = (const float*)d_in[0];
    const float* W        = (const float*)d_in[1];
    const int*   nodePtr  = (const int*)d_in[2];
    const int*   edgeList = (const int*)d_in[3];

    float* XW  = (float*)d_ws;
    float* out = (float*)d_out;

    // Stage 1: 12500 tile-waves, 8 waves (256 threads) per block.
    const int gemmBlocks = (GCN_TILES + 7) / 8;          // 1563
    gcn_gemm_wmma<<<gemmBlocks, 256, 0, stream>>>(X, W, XW);

    // Stage 2: one wave per node -> 50000 waves, 8 per block.
    const int aggBlocks = (GCN_N_NODES * 32 + 255) / 256; // 6250
    gcn_spmm_agg<<<aggBlocks, 256, 0, stream>>>(XW, nodePtr, edgeList, out);
}